// DTWLoss_12489764897117
// MI455X (gfx1250) — compile-verified
//
#include <hip/hip_runtime.h>
#include <hip/hip_bf16.h>
#include <math.h>

// ---------------------------------------------------------------------------
// DTW loss for MI455X (gfx1250, wave32).
//   out = 0.5 * mean|pred-target|  +  0.5 * DTW(pred[0], target[0]) / (S*F)
// B=16, S=2048, F=128.
//
// Pipeline:
//   k1: MAE block partial sums (bandwidth bound, 32MB)
//   k2: row norms xx[i]=|x_i|^2, yy[j]=|y_j|^2
//   k3: cost matrix d[i][j]=sqrt(max(xx+yy-2*x.y^T,1e-12)) via fp32 WMMA
//       (V_WMMA_F32_16X16X4_F32, one wave per 16x16 tile, K=128 -> 32 wmma)
//   k4: persistent single-workgroup DTW DP: per row a block cumsum scan and a
//       block exclusive-cummin scan (wave shfl scans + LDS cross-wave combine),
//       prev row in LDS; cost rows are L2-resident (16MB << 192MB L2).
// ---------------------------------------------------------------------------

#define S_LEN 2048
#define F_LEN 128
#define B_SZ 16
#define BIGF 3.0e38f

typedef float v2f __attribute__((ext_vector_type(2)));
typedef float v8f __attribute__((ext_vector_type(8)));

// ------------------------------ k1: MAE partials ---------------------------
__global__ void mae_partials_kernel(const float* __restrict__ p,
                                    const float* __restrict__ t,
                                    float* __restrict__ part, int n) {
    __shared__ float red[256];
    float s = 0.0f;
    for (int i = blockIdx.x * blockDim.x + threadIdx.x; i < n;
         i += gridDim.x * blockDim.x)
        s += fabsf(p[i] - t[i]);
    red[threadIdx.x] = s;
    __syncthreads();
    for (int o = 128; o > 0; o >>= 1) {
        if (threadIdx.x < o) red[threadIdx.x] += red[threadIdx.x + o];
        __syncthreads();
    }
    if (threadIdx.x == 0) part[blockIdx.x] = red[0];
}

// ------------------------------ k2: row norms ------------------------------
__global__ void norms_kernel(const float* __restrict__ x,
                             const float* __restrict__ y,
                             float* __restrict__ xx, float* __restrict__ yy) {
    int i = blockIdx.x * blockDim.x + threadIdx.x;
    if (i < S_LEN) {
        const float* xr = x + (size_t)i * F_LEN;
        const float* yr = y + (size_t)i * F_LEN;
        float sx = 0.0f, sy = 0.0f;
#pragma unroll 4
        for (int k = 0; k < F_LEN; ++k) {
            sx = fmaf(xr[k], xr[k], sx);
            sy = fmaf(yr[k], yr[k], sy);
        }
        xx[i] = sx;
        yy[i] = sy;
    }
}

// ------------------- k3: cost matrix via fp32 WMMA -------------------------
// One wave per 16x16 output tile. A = x-tile (16xK), B = y-tile^T (Kx16).
// f32 16x16x4: A lane L<16 holds {K0,K1} of row M=L, lane L>=16 holds {K2,K3};
// B mirrors with N in place of M. C/D: VGPR e -> M = e + 8*(lane>=16), N=lane&15.
__global__ __launch_bounds__(128) void cost_wmma_kernel(
    const float* __restrict__ x, const float* __restrict__ y,
    const float* __restrict__ xx, const float* __restrict__ yy,
    float* __restrict__ dmat) {
    int waveId = threadIdx.x >> 5;
    int lane = threadIdx.x & 31;
    int tile = blockIdx.x * 4 + waveId;        // 16384 tiles = 128 x 128
    int tm = tile >> 7, tn = tile & 127;
    int row = tm << 4, col = tn << 4;
    int half = lane >> 4, sub = lane & 15;

    const float* arow = x + (size_t)(row + sub) * F_LEN + 2 * half;
    const float* brow = y + (size_t)(col + sub) * F_LEN + 2 * half;

    v8f acc = {0.f, 0.f, 0.f, 0.f, 0.f, 0.f, 0.f, 0.f};
#pragma unroll 8
    for (int k = 0; k < F_LEN; k += 4) {
        v2f a = *(const v2f*)(arow + k);
        v2f b = *(const v2f*)(brow + k);
        acc = __builtin_amdgcn_wmma_f32_16x16x4_f32(
            /*neg_a=*/false, a, /*neg_b=*/false, b,
            /*c_mod=*/(short)0, acc, /*reuse_a=*/false, /*reuse_b=*/false);
    }

    float yn = yy[col + sub];
#pragma unroll
    for (int e = 0; e < 8; ++e) {
        int m = row + e + 8 * half;
        float sq = xx[m] + yn - 2.0f * acc[e];
        dmat[(size_t)m * S_LEN + col + sub] = sqrtf(fmaxf(sq, 1e-12f));
    }
}

// ------------------------- block scan primitives ---------------------------
__device__ __forceinline__ float wave_incl_scan_sum(float x, int lane) {
#pragma unroll
    for (int off = 1; off < 32; off <<= 1) {
        float n = __shfl_up(x, off, 32);
        if (lane >= off) x += n;
    }
    return x;
}

__device__ __forceinline__ float wave_incl_scan_min(float x, int lane) {
#pragma unroll
    for (int off = 1; off < 32; off <<= 1) {
        float n = __shfl_up(x, off, 32);
        if (lane >= off) x = fminf(x, n);
    }
    return x;
}

// Inclusive block-wide sum scan of per-thread aggregates (1024 thr, 32 waves).
__device__ __forceinline__ float block_incl_scan_sum(float v, float* lds32,
                                                     int tid) {
    int lane = tid & 31, wave = tid >> 5;
    float xi = wave_incl_scan_sum(v, lane);
    if (lane == 31) lds32[wave] = xi;
    __syncthreads();
    if (wave == 0) {
        float w = lds32[lane];
        w = wave_incl_scan_sum(w, lane);
        lds32[lane] = w;
    }
    __syncthreads();
    float off = (wave == 0) ? 0.0f : lds32[wave - 1];
    return xi + off;
}

// Exclusive block-wide min scan (identity BIGF) of per-thread aggregates.
__device__ __forceinline__ float block_excl_scan_min(float v, float* lds32,
                                                     int tid) {
    int lane = tid & 31, wave = tid >> 5;
    float xi = wave_incl_scan_min(v, lane);
    if (lane == 31) lds32[wave] = xi;
    __syncthreads();
    if (wave == 0) {
        float w = lds32[lane];
        w = wave_incl_scan_min(w, lane);
        lds32[lane] = w;
    }
    __syncthreads();
    float waveoff = (wave == 0) ? BIGF : lds32[wave - 1];
    float e = __shfl_up(xi, 1, 32);
    if (lane == 0) e = BIGF;
    return fminf(e, waveoff);
}

// ------------------------- k4: DTW DP + combine ----------------------------
// D[i,j] = d[i,j] + min(D[i-1,j], D[i,j-1], D[i-1,j-1])
//        = C_j + cummin_{k<=j}(b_k - C_k),  b = d + min(prev, shift(prev)),
//          C = cumsum(d row).  Thread t owns columns 2t, 2t+1.
__global__ __launch_bounds__(1024) void dtw_kernel(
    const float* __restrict__ dmat, const float* __restrict__ part,
    float* __restrict__ out) {
    __shared__ float prev[S_LEN];
    __shared__ float sA[32];   // sum-scan partials
    __shared__ float sB[32];   // min-scan partials
    __shared__ float red[1024];
    __shared__ float maeTot;

    int tid = threadIdx.x;
    int j0 = 2 * tid;

    // --- fold 4096 MAE partials (fixed order -> deterministic) ---
    float s = part[tid] + part[tid + 1024] + part[tid + 2048] + part[tid + 3072];
    red[tid] = s;
    __syncthreads();
    for (int o = 512; o > 0; o >>= 1) {
        if (tid < o) red[tid] += red[tid + o];
        __syncthreads();
    }
    if (tid == 0) maeTot = red[0];

    // --- row 0: D[0,:] = cumsum(d[0,:]) ---
    float d0 = dmat[j0], d1 = dmat[j0 + 1];
    float tsum = d0 + d1;
    float incl = block_incl_scan_sum(tsum, sA, tid);
    float excl = incl - tsum;
    prev[j0] = excl + d0;
    prev[j0 + 1] = incl;
    __syncthreads();

    // --- rows 1..S-1 ---
    for (int i = 1; i < S_LEN; ++i) {
        const float* drow = dmat + (size_t)i * S_LEN;
        d0 = drow[j0];
        d1 = drow[j0 + 1];
        if (i + 1 < S_LEN)  // warm next row toward L0/L2 while we scan
            __builtin_prefetch(drow + S_LEN + j0, 0, 0);

        float p0 = prev[j0];
        float p1 = prev[j0 + 1];
        float pm1 = (tid == 0) ? BIGF : prev[j0 - 1];
        float b0 = d0 + fminf(p0, pm1);  // d + min(D[i-1,j], D[i-1,j-1])
        float b1 = d1 + fminf(p1, p0);

        tsum = d0 + d1;
        incl = block_incl_scan_sum(tsum, sA, tid);  // barriers: prev reads done
        excl = incl - tsum;
        float c0 = excl + d0, c1 = incl;

        float t0 = b0 - c0, t1 = b1 - c1;
        float E = block_excl_scan_min(fminf(t0, t1), sB, tid);
        float cm0 = fminf(E, t0);
        float cm1 = fminf(E, fminf(t0, t1));

        prev[j0] = c0 + cm0;
        prev[j0 + 1] = c1 + cm1;
        __syncthreads();
    }

    if (tid == 0) {
        float mae = maeTot / (float)(B_SZ * S_LEN * F_LEN);
        float dtw = prev[S_LEN - 1] / (float)(S_LEN * F_LEN);
        out[0] = 0.5f * mae + 0.5f * dtw;
    }
}

// ------------------------------ launcher -----------------------------------
extern "C" void kernel_launch(void* const* d_in, const int* in_sizes, int n_in,
                              void* d_out, int out_size, void* d_ws,
                              size_t ws_size, hipStream_t stream) {
    const float* pred = (const float*)d_in[0];
    const float* target = (const float*)d_in[1];
    float* out = (float*)d_out;
    const int n = in_sizes[0];  // B*S*F = 4,194,304

    // workspace layout (floats)
    float* dmat = (float*)d_ws;                       // 2048*2048
    float* xx = dmat + (size_t)S_LEN * S_LEN;         // 2048
    float* yy = xx + S_LEN;                           // 2048
    float* partials = yy + S_LEN;                     // 4096

    // x = pred[0], y = target[0] (first S*F floats of each)
    const float* x = pred;
    const float* y = target;

    mae_partials_kernel<<<4096, 256, 0, stream>>>(pred, target, partials, n);
    norms_kernel<<<(S_LEN + 255) / 256, 256, 0, stream>>>(x, y, xx, yy);
    cost_wmma_kernel<<<4096, 128, 0, stream>>>(x, y, xx, yy, dmat);
    dtw_kernel<<<1, 1024, 0, stream>>>(dmat, partials, out);
}